// kernel_generated_0_21887153341190
// MI455X (gfx1250) — compile-verified
//
#include <hip/hip_runtime.h>

typedef __attribute__((ext_vector_type(2))) float v2f;
typedef __attribute__((ext_vector_type(8))) float v8f;

#define C_OUT   24
#define KTOT    72                      // 24 in-channels * 3 taps
#define HHALF   56
#define WIDTH   112
#define SPATIAL (HHALF * 2 * WIDTH)     // 12544 floats per channel plane

// One wave32 computes a 24(M, padded 32)x16(N) f32 tile: per K-step it loads
// one shared B fragment and feeds two v_wmma_f32_16x16x4_f32 (M-tiles 0 and 1).
__global__ void __launch_bounds__(256)
shiftconv_wmma(const float* __restrict__ x,
               const float* __restrict__ Wf,
               float* __restrict__ y)
{
    const int lane  = threadIdx.x & 31;
    const int sTile = (blockIdx.x * blockDim.x + threadIdx.x) >> 5;  // 0..783
    if (sTile >= 784) return;                    // wave-uniform guard

    const int row112 = sTile / 7;                // = 2n + o (output H row)
    const int mb     = (sTile - row112 * 7) << 4;
    const int n      = row112 >> 1;
    const int o      = row112 & 1;

    const int half = lane >> 4;                  // 0: lanes 0-15, 1: 16-31
    const int lp   = lane & 15;
    const int m    = mb + lp;                    // spatial column (pre-roll)

    // Roll(+1) on 56-dim => src half-row (n+k-2) mod 56; zero-pad masks at the
    // tap boundaries (valid iff 0 <= n+k-1 < 56). Invalid taps read a safe row.
    float tapMask[3];
    int   tapOff [3];
#pragma unroll
    for (int k = 0; k < 3; ++k) {
        int  q     = n + k - 1;
        bool valid = (q >= 0) && (q < HHALF);
        int  n2    = (q <= 0) ? (HHALF - 1) : (q - 1);   // always in-bounds
        tapMask[k] = valid ? 1.0f : 0.0f;
        tapOff[k]  = (n2 * 2 + o) * WIDTH + m;
    }

    // K index for fragment element e at step kk = 3t+p:  r = 12t + (4p + 2*half + e).
    // c = 4p + 2*half + e in [0,12) is t-invariant -> 6 base pointers + masks;
    // the t-loop then uses immediate offsets (4*t*SPATIAL floats <= ~1MB).
    const float* bPtr [3][2];
    float        bMask[3][2];
#pragma unroll
    for (int p = 0; p < 3; ++p) {
#pragma unroll
        for (int e = 0; e < 2; ++e) {
            int c  = 4 * p + 2 * half + e;
            int jo = c / 3;                      // in-channel offset
            int k  = c - 3 * jo;                 // tap
            bPtr [p][e] = x + jo * SPATIAL + tapOff[k];
            bMask[p][e] = tapMask[k];
        }
    }

    // A rows: tile0 = lp (always valid), tile1 = 16+lp (rows >= 24 zeroed).
    const bool   i1Valid = (16 + lp) < C_OUT;
    const float  aScale1 = i1Valid ? 1.0f : 0.0f;
    const float* aPtr0   = Wf + lp * KTOT + 2 * half;
    const float* aPtr1   = Wf + (i1Valid ? (16 + lp) : 0) * KTOT + 2 * half;

    v8f acc0 = {};
    v8f acc1 = {};

#pragma unroll
    for (int t = 0; t < 6; ++t) {
#pragma unroll
        for (int p = 0; p < 3; ++p) {
            const int aOff = 12 * t + 4 * p;         // + e, contiguous pair
            const int bOff = 4 * t * SPATIAL;        // immediate offset

            v2f a0, a1, b;
            a0.x = aPtr0[aOff + 0];
            a0.y = aPtr0[aOff + 1];
            a1.x = aScale1 * aPtr1[aOff + 0];
            a1.y = aScale1 * aPtr1[aOff + 1];
            b.x  = bMask[p][0] * bPtr[p][0][bOff];
            b.y  = bMask[p][1] * bPtr[p][1][bOff];

            acc0 = __builtin_amdgcn_wmma_f32_16x16x4_f32(
                false, a0, false, b, (short)0, acc0, false, false);
            acc1 = __builtin_amdgcn_wmma_f32_16x16x4_f32(
                false, a1, false, b, (short)0, acc1, false, false);
        }
    }

    // D layout: VGPR e -> M = e + 8*half, N = lp. Output roll on last dim:
    // out[i, 2n+o, m] -> y[i, 2n+o, (m+1) % 112].
    const int mOut = (m + 1 == WIDTH) ? 0 : (m + 1);
    const int base = row112 * WIDTH + mOut;

#pragma unroll
    for (int e = 0; e < 8; ++e) {
        int i0 = e + 8 * half;                       // 0..15, always valid
        y[i0 * SPATIAL + base] = acc0[e];
    }
    if (half == 0) {                                 // tile1 rows 16..23 only
#pragma unroll
        for (int e = 0; e < 8; ++e) {
            y[(16 + e) * SPATIAL + base] = acc1[e];
        }
    }
}

extern "C" void kernel_launch(void* const* d_in, const int* in_sizes, int n_in,
                              void* d_out, int out_size, void* d_ws, size_t ws_size,
                              hipStream_t stream) {
    (void)in_sizes; (void)n_in; (void)out_size; (void)d_ws; (void)ws_size;
    const float* x = (const float*)d_in[0];   // (1,24,112,112) f32
    const float* W = (const float*)d_in[1];   // (24,24,3) f32
    float*       y = (float*)d_out;           // (1,24,112,112) f32

    // 784 spatial tiles, one wave each = 98 blocks * 8 waves
    dim3 grid(98), block(256);
    hipLaunchKernelGGL(shiftconv_wmma, grid, block, 0, stream, x, W, y);
}